// RMAC_30734785970592
// MI455X (gfx1250) — compile-verified
//
#include <hip/hip_runtime.h>
#include <math.h>

#define AS1 __attribute__((address_space(1)))
#define AS3 __attribute__((address_space(3)))

typedef int v4i __attribute__((vector_size(16)));

#define EPSF 1e-6f
#define PPW   8      // planes per wave
#define WAVES 4      // waves per block
#define NPLANE 65536 // B*C = 32*2048

#if defined(__has_builtin)
#  if __has_builtin(__builtin_amdgcn_global_load_async_to_lds_b128)
#    define HAVE_ASYNC_BUILTIN 1
#  else
#    define HAVE_ASYNC_BUILTIN 0
#  endif
#  if __has_builtin(__builtin_amdgcn_s_wait_asynccnt)
#    define HAVE_WAIT_BUILTIN 1
#  else
#    define HAVE_WAIT_BUILTIN 0
#  endif
#else
#  define HAVE_ASYNC_BUILTIN 0
#  define HAVE_WAIT_BUILTIN 0
#endif

// Copy one 4KB plane (row-major 32x32 f32) from global to LDS, asynchronously.
// Each lane moves 16B per instruction; 8 instructions move the whole plane.
// INST_OFFSET is added to BOTH the global and the LDS address (ISA 08 §4.4).
__device__ __forceinline__ void async_plane(const char* g, char* l) {
#if HAVE_ASYNC_BUILTIN
  AS1 v4i* gp = (AS1 v4i*)g;
  AS3 v4i* lp = (AS3 v4i*)l;
  __builtin_amdgcn_global_load_async_to_lds_b128(gp, lp,    0, 0);
  __builtin_amdgcn_global_load_async_to_lds_b128(gp, lp,  512, 0);
  __builtin_amdgcn_global_load_async_to_lds_b128(gp, lp, 1024, 0);
  __builtin_amdgcn_global_load_async_to_lds_b128(gp, lp, 1536, 0);
  __builtin_amdgcn_global_load_async_to_lds_b128(gp, lp, 2048, 0);
  __builtin_amdgcn_global_load_async_to_lds_b128(gp, lp, 2560, 0);
  __builtin_amdgcn_global_load_async_to_lds_b128(gp, lp, 3072, 0);
  __builtin_amdgcn_global_load_async_to_lds_b128(gp, lp, 3584, 0);
#else
  unsigned laddr = (unsigned)(size_t)(AS3 void*)l;
  asm volatile(
      "global_load_async_to_lds_b128 %0, %1, off\n\t"
      "global_load_async_to_lds_b128 %0, %1, off offset:512\n\t"
      "global_load_async_to_lds_b128 %0, %1, off offset:1024\n\t"
      "global_load_async_to_lds_b128 %0, %1, off offset:1536\n\t"
      "global_load_async_to_lds_b128 %0, %1, off offset:2048\n\t"
      "global_load_async_to_lds_b128 %0, %1, off offset:2560\n\t"
      "global_load_async_to_lds_b128 %0, %1, off offset:3072\n\t"
      "global_load_async_to_lds_b128 %0, %1, off offset:3584"
      :: "v"(laddr), "v"(g) : "memory");
#endif
}

__device__ __forceinline__ void wait_async8() {
#if HAVE_WAIT_BUILTIN
  __builtin_amdgcn_s_wait_asynccnt(8);
#else
  asm volatile("s_wait_asynccnt 8" ::: "memory");
#endif
  asm volatile("" ::: "memory");
}
__device__ __forceinline__ void wait_async0() {
#if HAVE_WAIT_BUILTIN
  __builtin_amdgcn_s_wait_asynccnt(0);
#else
  asm volatile("s_wait_asynccnt 0" ::: "memory");
#endif
  asm volatile("" ::: "memory");
}

// Kernel 1: per (b,c) plane compute the 14 distinct region maxes.
// rm layout: rm[region][plane], region 0 = full 32x32 (weighted 2x in kernel 2).
__global__ void __launch_bounds__(128) rmac_regmax(const float* __restrict__ x,
                                                   float* __restrict__ rm) {
  __shared__ float buf[WAVES][2][1024]; // double buffer: one 4KB plane each
  __shared__ float scr[WAVES][256];     // 6 segments x 32 cols (+ padding)
  const int wave = threadIdx.x >> 5;
  const int lane = threadIdx.x & 31;
  const int pbase = (blockIdx.x * WAVES + wave) * PPW;

  const char* g = (const char*)x + (size_t)pbase * 4096 + lane * 16;
  char* l0 = (char*)&buf[wave][0][0] + lane * 16;
  char* l1 = (char*)&buf[wave][1][0] + lane * 16;

  async_plane(g, l0); // prefetch plane 0

  for (int k = 0; k < PPW; ++k) {
    if (k + 1 < PPW) {
      async_plane(g + (size_t)(k + 1) * 4096, (k & 1) ? l0 : l1);
      wait_async8(); // plane k's 8 ops done; k+1's 8 still in flight
    } else {
      wait_async0();
    }

    // lane = column; read 32 rows of this column (bank-conflict-free b32 reads)
    const float* P = &buf[wave][k & 1][0];
    float v[32];
#pragma unroll
    for (int r = 0; r < 32; ++r) v[r] = P[r * 32 + lane];

    // row-segment maxes via 8-row block tree
    float b0 = v[0], b1 = v[8], b2 = v[16], b3 = v[24];
#pragma unroll
    for (int r = 1; r < 8; ++r) {
      b0 = fmaxf(b0, v[r]);       b1 = fmaxf(b1, v[8 + r]);
      b2 = fmaxf(b2, v[16 + r]);  b3 = fmaxf(b3, v[24 + r]);
    }
    float m16 = v[16]; // rows 16..20
#pragma unroll
    for (int r = 17; r <= 20; ++r) m16 = fmaxf(m16, v[r]);
    float m11 = v[11]; // rows 11..15
#pragma unroll
    for (int r = 12; r <= 15; ++r) m11 = fmaxf(m11, v[r]);

    float s3 = fmaxf(b0, b1);   // rows 0..15
    float s5 = fmaxf(b2, b3);   // rows 16..31
    float s4 = fmaxf(b1, b2);   // rows 8..23
    float s0 = fmaxf(s3, s5);   // rows 0..31
    float s1 = fmaxf(s3, m16);  // rows 0..20
    float s2 = fmaxf(m11, s5);  // rows 11..31

    scr[wave][  0 + lane] = s0;
    scr[wave][ 32 + lane] = s1;
    scr[wave][ 64 + lane] = s2;
    scr[wave][ 96 + lane] = s3;
    scr[wave][128 + lane] = s4;
    scr[wave][160 + lane] = s5;
    __builtin_amdgcn_wave_barrier(); // LDS ops are in-order within a wave

    // lanes 0..13 each reduce one region's column range
    int seg, cst, cnt;
    if (lane == 0)      { seg = 0; cst = 0; cnt = 32; }
    else if (lane < 5)  { int q = lane - 1; seg = 1 + (q >> 1); cst = (q & 1) * 11; cnt = 21; }
    else if (lane < 14) { int q = lane - 5; int qa = q / 3; seg = 3 + qa; cst = (q - qa * 3) * 8; cnt = 16; }
    else                { seg = 0; cst = 0; cnt = 0; }

    const float* S = &scr[wave][seg * 32 + cst];
    float m = -3.402823466e38f;
#pragma unroll
    for (int i = 0; i < 32; ++i)
      if (i < cnt) m = fmaxf(m, S[i]);

    if (lane < 14) rm[lane * NPLANE + (pbase + k)] = m;
    __builtin_amdgcn_wave_barrier();
  }
}

// Kernel 2: per batch b, for each region L2-normalize over C and accumulate.
__global__ void __launch_bounds__(256) rmac_norm(const float* __restrict__ rm,
                                                 float* __restrict__ out) {
  const int b = blockIdx.x;
  const int t = threadIdx.x;
  __shared__ float red[8];
  float acc[8] = {0.f, 0.f, 0.f, 0.f, 0.f, 0.f, 0.f, 0.f};

  for (int r = 0; r < 14; ++r) {
    float m[8];
    float ss = 0.f;
#pragma unroll
    for (int j = 0; j < 8; ++j) {
      m[j] = rm[r * NPLANE + b * 2048 + j * 256 + t];
      ss += m[j] * m[j];
    }
#pragma unroll
    for (int o = 16; o >= 1; o >>= 1) ss += __shfl_xor(ss, o, 32);
    __syncthreads(); // protect red[] from previous iteration's readers
    if ((t & 31) == 0) red[t >> 5] = ss;
    __syncthreads();
    float tot = 0.f;
#pragma unroll
    for (int w2 = 0; w2 < 8; ++w2) tot += red[w2];
    float sc = ((r == 0) ? 2.0f : 1.0f) / (sqrtf(tot) + EPSF);
#pragma unroll
    for (int j = 0; j < 8; ++j) acc[j] += sc * m[j];
  }
#pragma unroll
  for (int j = 0; j < 8; ++j) out[b * 2048 + j * 256 + t] = acc[j];
}

extern "C" void kernel_launch(void* const* d_in, const int* in_sizes, int n_in,
                              void* d_out, int out_size, void* d_ws, size_t ws_size,
                              hipStream_t stream) {
  (void)in_sizes; (void)n_in; (void)out_size; (void)ws_size;
  const float* x = (const float*)d_in[0];
  float* rm = (float*)d_ws;                 // 14 * 65536 floats = 3.67 MB
  float* out = (float*)d_out;               // 65536 floats

  // 65536 planes / (4 waves * 8 planes) = 2048 blocks
  rmac_regmax<<<2048, 128, 0, stream>>>(x, rm);
  rmac_norm<<<32, 256, 0, stream>>>(rm, out);
}